// AdvancedParameterDistributionDPOLoss_6493990552354
// MI455X (gfx1250) — compile-verified
//
#include <hip/hip_runtime.h>
#include <math.h>

#define EPSV 1e-8f

typedef float v2f __attribute__((ext_vector_type(2)));
typedef float v8f __attribute__((ext_vector_type(8)));

// Workspace layout (floats). 8 tensors: t=0..3 current rows, t=4..7 initial rows.
enum {
  WS_SUM  = 0,   // [8] running sum
  WS_SQ   = 8,   // [8] running sum of squares
  WS_MAX  = 16,  // [8] running max
  WS_MEAN = 24,  // [8]
  WS_RSTD = 32,  // [8] 1/(std+eps)
  WS_ZMAX = 40,  // [8] (max-mean)*rstd
  WS_Z    = 48,  // [8] sum exp(z - zmax)
  WS_S    = 56,  // [8] sum e*(z - zmax)
  WS_CROSS= 64,  // [4] sum e_init * log(p_cur + eps)
  WS_TOTAL= 68
};

// ---------------------------------------------------------------------------
// Layout-independent wave-level sum of a 256-float LDS array using f32 WMMA.
// Must be called by ALL 32 lanes of wave 0 (EXEC all ones, required by WMMA).
// Each lane packs 8 values into the 2 A-matrix VGPR slots; with B == all-ones
// and C == 0, D[m][n] = sum_k A[m][k]. Summing the 8 D VGPRs per lane and
// adding the opposite half-wave (shfl_xor 16) gives the sum of every A slot.
// ---------------------------------------------------------------------------
__device__ __forceinline__ float wave0_wmma_sum256(const float* red) {
  const int l = threadIdx.x;  // 0..31
  float x0 = (red[l      ] + red[l +  32]) + (red[l +  64] + red[l +  96]);
  float x1 = (red[l + 128] + red[l + 160]) + (red[l + 192] + red[l + 224]);
  v2f a; a[0] = x0;   a[1] = x1;
  v2f b; b[0] = 1.0f; b[1] = 1.0f;
  v8f c = {};
  v8f d = __builtin_amdgcn_wmma_f32_16x16x4_f32(false, a, false, b,
                                                (short)0, c, false, false);
  float t = ((d[0] + d[1]) + (d[2] + d[3])) + ((d[4] + d[5]) + (d[6] + d[7]));
  t += __shfl_xor(t, 16, 32);
  return t;  // full sum of all 256 inputs, valid in every lane of wave 0
}

__device__ __forceinline__ void atomicMaxFloat(float* addr, float val) {
  unsigned int* ua = (unsigned int*)addr;
  unsigned int cur = *ua;
  while (__uint_as_float(cur) < val) {
    unsigned int prev = atomicCAS(ua, cur, __float_as_uint(val));
    if (prev == cur) break;
    cur = prev;
  }
}

// ---------------------------------------------------------------------------
// Pass 0: initialize workspace accumulators (workspace is poisoned by harness).
// ---------------------------------------------------------------------------
__global__ void init_ws_kernel(float* __restrict__ ws) {
  int i = threadIdx.x;
  if (i < WS_TOTAL)
    ws[i] = (i >= WS_MAX && i < WS_MAX + 8) ? -INFINITY : 0.0f;
}

// ---------------------------------------------------------------------------
// Pass 1: sum / sumsq / max for tensors {p, 4+p}. Streams 134 MB (fills L2).
// ---------------------------------------------------------------------------
__global__ void stats_kernel(const float* __restrict__ xc,
                             const float* __restrict__ xi,
                             long long n4, float* __restrict__ ws, int p) {
  const int t = (blockIdx.y == 0) ? p : (4 + p);
  const float4* __restrict__ X =
      (const float4*)((blockIdx.y == 0) ? xc : xi);
  const long long stride = (long long)gridDim.x * blockDim.x;
  float s = 0.0f, sq = 0.0f, mx = -INFINITY;
  for (long long i = (long long)blockIdx.x * blockDim.x + threadIdx.x;
       i < n4; i += stride) {
    __builtin_prefetch(&X[i + 2 * stride], 0, 3);  // global_prefetch_b8
    float4 v = X[i];
    s  += (v.x + v.y) + (v.z + v.w);
    sq += (v.x * v.x + v.y * v.y) + (v.z * v.z + v.w * v.w);
    mx  = fmaxf(mx, fmaxf(fmaxf(v.x, v.y), fmaxf(v.z, v.w)));
  }
  __shared__ float red[3][256];
  const int tid = threadIdx.x;
  red[0][tid] = s; red[1][tid] = sq; red[2][tid] = mx;
  __syncthreads();
  if (tid < 32) {
    float ts = wave0_wmma_sum256(red[0]);
    float tq = wave0_wmma_sum256(red[1]);
    float m = red[2][tid];
    #pragma unroll
    for (int k = 32; k < 256; k += 32) m = fmaxf(m, red[2][tid + k]);
    #pragma unroll
    for (int off = 16; off; off >>= 1) m = fmaxf(m, __shfl_xor(m, off, 32));
    if (tid == 0) {
      atomicAdd(&ws[WS_SUM + t], ts);
      atomicAdd(&ws[WS_SQ  + t], tq);
      atomicMaxFloat(&ws[WS_MAX + t], m);
    }
  }
}

// mean / rstd / zmax for tensors {p, 4+p}
__global__ void finalize_stats_kernel(float* __restrict__ ws, float n, int p) {
  int tid = threadIdx.x;
  if (tid < 2) {
    int t = (tid == 0) ? p : (4 + p);
    float sum = ws[WS_SUM + t], sumsq = ws[WS_SQ + t];
    float mean = sum / n;
    float var  = fmaxf(0.0f, (sumsq - sum * sum / n) / (n - 1.0f));  // ddof=1
    float rstd = 1.0f / (sqrtf(var) + EPSV);
    ws[WS_MEAN + t] = mean;
    ws[WS_RSTD + t] = rstd;
    ws[WS_ZMAX + t] = (ws[WS_MAX + t] - mean) * rstd;
  }
}

// ---------------------------------------------------------------------------
// Pass 2: Z = sum exp(z - zmax), S = sum e*(z - zmax). Re-reads from L2.
// ---------------------------------------------------------------------------
__global__ void expsum_kernel(const float* __restrict__ xc,
                              const float* __restrict__ xi,
                              long long n4, float* __restrict__ ws, int p) {
  const int t = (blockIdx.y == 0) ? p : (4 + p);
  const float4* __restrict__ X =
      (const float4*)((blockIdx.y == 0) ? xc : xi);
  const float mean = ws[WS_MEAN + t];
  const float rstd = ws[WS_RSTD + t];
  const float zmax = ws[WS_ZMAX + t];
  const long long stride = (long long)gridDim.x * blockDim.x;
  float Z = 0.0f, S = 0.0f;
  for (long long i = (long long)blockIdx.x * blockDim.x + threadIdx.x;
       i < n4; i += stride) {
    __builtin_prefetch(&X[i + 2 * stride], 0, 3);
    float4 v = X[i];
    float u0 = (v.x - mean) * rstd - zmax;
    float u1 = (v.y - mean) * rstd - zmax;
    float u2 = (v.z - mean) * rstd - zmax;
    float u3 = (v.w - mean) * rstd - zmax;
    float e0 = expf(u0), e1 = expf(u1), e2 = expf(u2), e3 = expf(u3);
    Z += (e0 + e1) + (e2 + e3);
    S += (e0 * u0 + e1 * u1) + (e2 * u2 + e3 * u3);
  }
  __shared__ float red[2][256];
  const int tid = threadIdx.x;
  red[0][tid] = Z; red[1][tid] = S;
  __syncthreads();
  if (tid < 32) {
    float tZ = wave0_wmma_sum256(red[0]);
    float tS = wave0_wmma_sum256(red[1]);
    if (tid == 0) {
      atomicAdd(&ws[WS_Z + t], tZ);
      atomicAdd(&ws[WS_S + t], tS);
    }
  }
}

// ---------------------------------------------------------------------------
// Pass 3: cross = sum e_init * log(p_cur + eps). Re-reads both from L2.
// ---------------------------------------------------------------------------
__global__ void cross_kernel(const float* __restrict__ xc,
                             const float* __restrict__ xi,
                             long long n4, float* __restrict__ ws, int p) {
  const float mc = ws[WS_MEAN + p],     rc = ws[WS_RSTD + p],
              zc = ws[WS_ZMAX + p];
  const float mi = ws[WS_MEAN + 4 + p], ri = ws[WS_RSTD + 4 + p],
              zi = ws[WS_ZMAX + 4 + p];
  const float invZc = 1.0f / ws[WS_Z + p];
  const float4* __restrict__ C4 = (const float4*)xc;
  const float4* __restrict__ I4 = (const float4*)xi;
  const long long stride = (long long)gridDim.x * blockDim.x;
  float acc = 0.0f;
  for (long long i = (long long)blockIdx.x * blockDim.x + threadIdx.x;
       i < n4; i += stride) {
    __builtin_prefetch(&C4[i + 2 * stride], 0, 3);
    __builtin_prefetch(&I4[i + 2 * stride], 0, 3);
    float4 cv = C4[i];
    float4 iv = I4[i];
    float uc0 = (cv.x - mc) * rc - zc, ui0 = (iv.x - mi) * ri - zi;
    float uc1 = (cv.y - mc) * rc - zc, ui1 = (iv.y - mi) * ri - zi;
    float uc2 = (cv.z - mc) * rc - zc, ui2 = (iv.z - mi) * ri - zi;
    float uc3 = (cv.w - mc) * rc - zc, ui3 = (iv.w - mi) * ri - zi;
    acc += expf(ui0) * logf(expf(uc0) * invZc + EPSV);
    acc += expf(ui1) * logf(expf(uc1) * invZc + EPSV);
    acc += expf(ui2) * logf(expf(uc2) * invZc + EPSV);
    acc += expf(ui3) * logf(expf(uc3) * invZc + EPSV);
  }
  __shared__ float red[256];
  const int tid = threadIdx.x;
  red[tid] = acc;
  __syncthreads();
  if (tid < 32) {
    float t = wave0_wmma_sum256(red);
    if (tid == 0) atomicAdd(&ws[WS_CROSS + p], t);
  }
}

// reward = -(1/P) * sum_p [ S_i/Z_i - log Z_i - cross_p/Z_i ]
__global__ void final_kernel(const float* __restrict__ ws,
                             float* __restrict__ out) {
  if (threadIdx.x == 0 && blockIdx.x == 0) {
    float tot = 0.0f;
    for (int p = 0; p < 4; ++p) {
      float Zi = ws[WS_Z + 4 + p];
      float kl = ws[WS_S + 4 + p] / Zi - logf(Zi) - ws[WS_CROSS + p] / Zi;
      tot += kl;
    }
    out[0] = -(tot * 0.25f);
  }
}

extern "C" void kernel_launch(void* const* d_in, const int* in_sizes, int n_in,
                              void* d_out, int out_size, void* d_ws,
                              size_t ws_size, hipStream_t stream) {
  const float* cur  = (const float*)d_in[0];  // [4, N] fp32
  const float* init = (const float*)d_in[1];  // [4, N] fp32
  const long long total = (long long)in_sizes[0];
  const long long N  = total / 4;   // elements per parameter tensor
  const long long n4 = N / 4;       // float4 count (N = 4096*4096, %4 == 0)
  float* ws  = (float*)d_ws;
  float* out = (float*)d_out;

  init_ws_kernel<<<1, 128, 0, stream>>>(ws);

  const int B = 2048;  // blocks per tensor per pass
  // Process one parameter pair (134 MB) at a time so passes 2/3 hit the
  // 192 MB L2 instead of re-pulling from HBM.
  for (int p = 0; p < 4; ++p) {
    const float* xc = cur  + (long long)p * N;
    const float* xi = init + (long long)p * N;
    stats_kernel   <<<dim3(B, 2), 256, 0, stream>>>(xc, xi, n4, ws, p);
    finalize_stats_kernel<<<1, 32, 0, stream>>>(ws, (float)N, p);
    expsum_kernel  <<<dim3(B, 2), 256, 0, stream>>>(xc, xi, n4, ws, p);
    cross_kernel   <<<dim3(B, 1), 256, 0, stream>>>(xc, xi, n4, ws, p);
  }
  final_kernel<<<1, 32, 0, stream>>>(ws, out);
}